// LevelSetLayer_11536282157683
// MI455X (gfx1250) — compile-verified
//
#include <hip/hip_runtime.h>

// 0-dim sublevel-set persistence (lower-star filtration) on a 128x128 grid.
// Single workgroup (1024 threads = 32 wave32 waves on one WGP):
//   Phase 0: stage f[16384] into LDS via CDNA5 async global->LDS mover
//            (global_load_async_to_lds_b128, ASYNCcnt) when available;
//            init parent[16384] in LDS. 128 KB dynamic LDS total.
//   Phase 1: build sortable 64-bit keys (float->monotonic u32 | edge index)
//            into the global workspace (padded to 65536).
//   Phase 2: in-place bitonic sort of the 65536 keys (136 barrier-separated
//            passes). Keys are unique -> deterministic, matches stable argsort.
//   Phase 3: serial Kruskal/union-find sweep by lane 0 against LDS-resident
//            parent[] (path halving + elder rule, exactly as the reference),
//            writing (birth,death) f-values per sorted edge, 0 for non-merges.

constexpr int NVERT = 128 * 128;   // 16384 vertices
constexpr int NPAD  = 65536;       // edges padded to power of two for bitonic
constexpr int NT    = 1024;        // one workgroup

#if __has_builtin(__builtin_amdgcn_global_load_async_to_lds_b128)
#define HAVE_ASYNC_LDS 1
#else
#define HAVE_ASYNC_LDS 0
#endif

typedef int v4i __attribute__((ext_vector_type(4)));

__global__ __launch_bounds__(NT) void persistence0_kernel(
    const float* __restrict__ f,
    const int*   __restrict__ edges,   // [E][2] int32
    float*       __restrict__ out,     // [E][2] float32
    unsigned long long* __restrict__ keys,  // workspace: NPAD u64
    int E)
{
    extern __shared__ unsigned char smem[];
    float* fs     = reinterpret_cast<float*>(smem);                         // 64 KB
    int*   parent = reinterpret_cast<int*>(smem + NVERT * sizeof(float));   // 64 KB

    const int tid = threadIdx.x;

    // ---- Phase 0: stage f into LDS, init union-find parents in LDS ----
#if HAVE_ASYNC_LDS
    {
        // Builtin signature (from hipcc diagnostic): AS1 int4* global src,
        // AS3 int4* LDS dst, imm offset, imm cpol.
        typedef __attribute__((address_space(1))) v4i gv4i;
        typedef __attribute__((address_space(3))) v4i lv4i;
        // 16 bytes per lane per issue; 4 issues per thread cover 64 KB.
        for (int i = tid * 4; i < NVERT; i += NT * 4) {
            __builtin_amdgcn_global_load_async_to_lds_b128(
                (gv4i*)(f + i), (lv4i*)(fs + i), /*offset=*/0, /*cpol=*/0);
        }
    }
    for (int i = tid; i < NVERT; i += NT) {
        parent[i] = i;
    }
    // Complete this wave's async global->LDS transfers before the barrier.
#if __has_builtin(__builtin_amdgcn_s_wait_asynccnt)
    __builtin_amdgcn_s_wait_asynccnt(0);
#else
    asm volatile("s_wait_asynccnt 0" ::: "memory");
#endif
#else
    for (int i = tid; i < NVERT; i += NT) {
        fs[i]     = f[i];
        parent[i] = i;
    }
#endif
    __syncthreads();

    // ---- Phase 1: build sort keys ----
    // key = monotonic(u32(ev)) in high 32 bits, edge index in low 32 bits.
    // Unique keys => bitonic sort result == stable argsort of ev.
    for (int e = tid; e < NPAD; e += NT) {
        // gfx1250 prefetch path (global_prefetch_b8): pull next edge chunk.
        __builtin_prefetch(edges + 2 * (e + NT), 0, 1);

        unsigned long long kk;
        if (e < E) {
            const int u = edges[2 * e + 0];
            const int v = edges[2 * e + 1];
            const float ev = fmaxf(fs[u], fs[v]);
            unsigned k = __float_as_uint(ev);
            // IEEE float -> order-preserving unsigned
            k ^= (k & 0x80000000u) ? 0xFFFFFFFFu : 0x80000000u;
            kk = (static_cast<unsigned long long>(k) << 32) |
                 static_cast<unsigned>(e);
        } else {
            // Pad entries: high word 0xFFFFFFFF sorts strictly after any
            // finite float key (max finite maps to 0xFF7FFFFF).
            kk = (0xFFFFFFFFull << 32) | static_cast<unsigned>(e);
        }
        keys[e] = kk;
    }
    __syncthreads();

    // ---- Phase 2: bitonic sort of keys[0..NPAD) in workspace ----
    for (unsigned k = 2; k <= (unsigned)NPAD; k <<= 1) {
        for (unsigned j = k >> 1; j > 0; j >>= 1) {
            for (unsigned i = tid; i < (unsigned)NPAD; i += NT) {
                const unsigned l = i ^ j;
                if (l > i) {
                    const unsigned long long a = keys[i];
                    const unsigned long long b = keys[l];
                    const bool up = ((i & k) == 0);
                    if (up ? (a > b) : (a < b)) {
                        keys[i] = b;
                        keys[l] = a;
                    }
                }
            }
            __syncthreads();
        }
    }

    // ---- Phase 3: serial Kruskal/union-find sweep (lane 0) ----
    // parent[] and fs[] are LDS-resident: the dependent-load find() chains
    // run at LDS latency instead of L2 latency.
    if (tid == 0) {
        for (int i = 0; i < E; ++i) {
            const unsigned e = static_cast<unsigned>(keys[i] & 0xFFFFFFFFull);
            const int u = edges[2 * e + 0];
            const int v = edges[2 * e + 1];

            // find with path halving (matches reference _find)
            int ru = u;
            for (;;) {
                const int p = parent[ru];
                if (p == ru) break;
                const int gp = parent[p];
                parent[ru] = gp;
                ru = gp;
            }
            int rv = v;
            for (;;) {
                const int p = parent[rv];
                if (p == rv) break;
                const int gp = parent[p];
                parent[rv] = gp;
                rv = gp;
            }

            float birth = 0.0f, death = 0.0f;
            if (ru != rv) {
                // elder rule: root is its component's argmin vertex
                const bool u_elder = fs[ru] <= fs[rv];
                const int elder   = u_elder ? ru : rv;
                const int younger = u_elder ? rv : ru;
                const int dvtx    = (fs[u] >= fs[v]) ? u : v;
                parent[younger] = elder;
                birth = fs[younger];
                death = fs[dvtx];
            }
            out[2 * i + 0] = birth;
            out[2 * i + 1] = death;
        }
    }
}

extern "C" void kernel_launch(void* const* d_in, const int* in_sizes, int n_in,
                              void* d_out, int out_size, void* d_ws, size_t ws_size,
                              hipStream_t stream) {
    (void)n_in; (void)out_size; (void)ws_size;

    const float* f     = reinterpret_cast<const float*>(d_in[0]);
    const int*   edges = reinterpret_cast<const int*>(d_in[1]);
    float*       out   = reinterpret_cast<float*>(d_out);
    unsigned long long* keys = reinterpret_cast<unsigned long long*>(d_ws); // 512 KB

    const int E = in_sizes[1] / 2;

    // 128 KB dynamic LDS (fs + parent) — fits CDNA5's WGP LDS.
    const size_t shmem = static_cast<size_t>(NVERT) * (sizeof(float) + sizeof(int));

    persistence0_kernel<<<1, NT, shmem, stream>>>(f, edges, out, keys, E);
}